// KuramotoModule_7189775253867
// MI455X (gfx1250) — compile-verified
//
#include <hip/hip_runtime.h>
#include <math.h>

typedef float v2f __attribute__((ext_vector_type(2)));
typedef float v8f __attribute__((ext_vector_type(8)));

#define DIM   1024
#define SEQ   2048
#define BATCH 8
#define NSTEP 5

#define PI_F      3.14159265358979323846f
#define TWO_PI_F  6.28318530717958647692f
#define INV_2PI_F 0.15915494309189533577f
#define INV_PHI_F 0.61803398874989484820f
#define DT_F      0.01f

// ---------------------------------------------------------------------------
// Kernel 1: fused dual projection via fp32 WMMA.
//   A (32 tokens x 1024) * B (1024 x 16): B col0 = W_phase, col1 = W_freq,
//   cols 2..15 read a zeroed LDS row (no EXEC divergence in the hot loop).
//   One wave handles two 16-row WMMA tiles sharing each B fragment.
// ---------------------------------------------------------------------------
__global__ void __launch_bounds__(256)
kuramoto_proj_wmma(const float* __restrict__ h,
                   const float* __restrict__ Wp,
                   const float* __restrict__ bp,
                   const float* __restrict__ Wf,
                   const float* __restrict__ bf,
                   float* __restrict__ phases,
                   float* __restrict__ freqs) {
    // rows: [0]=W_phase, [1]=W_freq, [2]=zeros (read by B columns 2..15)
    __shared__ float wlds[3 * DIM];
    for (int i = threadIdx.x; i < DIM; i += blockDim.x) {
        wlds[i]           = Wp[i];
        wlds[DIM + i]     = Wf[i];
        wlds[2 * DIM + i] = 0.0f;
    }
    __syncthreads();

    const int lane = threadIdx.x & 31;
    const int wave = threadIdx.x >> 5;
    const int tile = blockIdx.x * (blockDim.x >> 5) + wave;  // 32 tokens/wave
    const int m    = lane & 15;   // A-row (within 16) / B-column for this lane
    const int half = lane >> 4;   // K sub-pair select per ISA 16x4 fp32 layout
    const int koff = 2 * half;

    const float* __restrict__ row0 = h + (size_t)(tile * 32 + m) * DIM;
    const float* __restrict__ row1 = row0 + (size_t)16 * DIM;
    // branch-free B address: columns >=2 stream the zero row
    const int wbase = ((m < 2) ? m : 2) * DIM + koff;

    v8f acc0 = {0.f, 0.f, 0.f, 0.f, 0.f, 0.f, 0.f, 0.f};
    v8f acc1 = {0.f, 0.f, 0.f, 0.f, 0.f, 0.f, 0.f, 0.f};

    #pragma unroll 4
    for (int k = 0; k < DIM; k += 4) {
        v2f a0 = *(const v2f*)(row0 + k + koff);      // global_load_b64
        v2f a1 = *(const v2f*)(row1 + k + koff);      // global_load_b64
        v2f b  = *(const v2f*)(&wlds[wbase + k]);     // ds_load_b64 (shared)
        acc0 = __builtin_amdgcn_wmma_f32_16x16x4_f32(
            false, a0, false, b, (short)0, acc0, false, false);
        acc1 = __builtin_amdgcn_wmma_f32_16x16x4_f32(
            false, a1, false, b, (short)0, acc1, false, false);
    }

    // D layout (16x16 f32): lane n<16 holds rows 0..7 in acc[0..7],
    // lane n+16 holds rows 8..15.  Column 0 -> phases, column 1 -> freqs.
    if (m < 2) {
        const float bias = (m == 0) ? bp[0] : bf[0];
        #pragma unroll
        for (int v = 0; v < 8; ++v) {
            const int mm     = v + half * 8;
            const int token0 = tile * 32 + mm;        // first 16-row tile
            const int token1 = token0 + 16;           // second 16-row tile
            const float val0 = acc0[v] + bias;
            const float val1 = acc1[v] + bias;
            if (m == 0) {
                phases[token0] = tanhf(val0) * PI_F;
                phases[token1] = tanhf(val1) * PI_F;
            } else {
                freqs[token0] = val0 + (float)(token0 & (SEQ - 1)) * INV_PHI_F;
                freqs[token1] = val1 + (float)(token1 & (SEQ - 1)) * INV_PHI_F;
            }
        }
    }
}

// ---------------------------------------------------------------------------
// Kernel 2: Kuramoto dynamics.  sum_j sin(ti - tj) = sin(ti)*C - cos(ti)*S
// with C = sum cos, S = sum sin.  One block per batch row, phases in regs.
// ---------------------------------------------------------------------------
__global__ void __launch_bounds__(512)
kuramoto_dynamics(const float* __restrict__ phases_in,
                  const float* __restrict__ freqs,
                  const float* __restrict__ coupling_scale,
                  float* __restrict__ out_phases,
                  float* __restrict__ out_r) {
    const int b   = blockIdx.x;
    const int tid = threadIdx.x;           // 512 threads
    const int PER = SEQ / 512;             // 4 phases per thread

    __shared__ float rc[512];
    __shared__ float rs[512];

    float th[PER], fr[PER];
    #pragma unroll
    for (int i = 0; i < PER; ++i) {
        const int idx = tid * PER + i;
        th[i] = phases_in[b * SEQ + idx];
        fr[i] = freqs[b * SEQ + idx];
    }

    const float K      = 1.0f / (1.0f + expf(-coupling_scale[0]));  // COUPLING=1
    const float KoverN = K / (float)SEQ;

    for (int step = 0; step < NSTEP; ++step) {
        float cs = 0.f, ss = 0.f;
        float sv[PER], cv[PER];
        #pragma unroll
        for (int i = 0; i < PER; ++i) {
            sv[i] = sinf(th[i]);
            cv[i] = cosf(th[i]);
            cs += cv[i];
            ss += sv[i];
        }
        rc[tid] = cs; rs[tid] = ss;
        __syncthreads();
        for (int off = 256; off > 0; off >>= 1) {
            if (tid < off) { rc[tid] += rc[tid + off]; rs[tid] += rs[tid + off]; }
            __syncthreads();
        }
        const float Csum = rc[0];
        const float Ssum = rs[0];
        __syncthreads();   // all reads of rc[0]/rs[0] done before next overwrite

        #pragma unroll
        for (int i = 0; i < PER; ++i) {
            const float d = fr[i] + KoverN * (sv[i] * Csum - cv[i] * Ssum);
            float x = th[i] + DT_F * d + PI_F;
            x = x - TWO_PI_F * floorf(x * INV_2PI_F);   // remainder -> [0, 2pi)
            th[i] = x - PI_F;
        }
    }

    // order parameter r = |mean exp(i*theta)|
    float cs = 0.f, ss = 0.f;
    #pragma unroll
    for (int i = 0; i < PER; ++i) {
        cs += cosf(th[i]);
        ss += sinf(th[i]);
        out_phases[b * SEQ + tid * PER + i] = th[i];
    }
    rc[tid] = cs; rs[tid] = ss;
    __syncthreads();
    for (int off = 256; off > 0; off >>= 1) {
        if (tid < off) { rc[tid] += rc[tid + off]; rs[tid] += rs[tid + off]; }
        __syncthreads();
    }
    if (tid == 0) {
        const float c = rc[0] / (float)SEQ;
        const float s = rs[0] / (float)SEQ;
        out_r[b] = sqrtf(c * c + s * s);
    }
}

// ---------------------------------------------------------------------------
extern "C" void kernel_launch(void* const* d_in, const int* in_sizes, int n_in,
                              void* d_out, int out_size, void* d_ws, size_t ws_size,
                              hipStream_t stream) {
    const float* h  = (const float*)d_in[0];   // (8, 2048, 1024)
    const float* Wp = (const float*)d_in[1];   // (1024, 1)
    const float* bp = (const float*)d_in[2];   // (1,)
    const float* Wf = (const float*)d_in[3];   // (1024, 1)
    const float* bf = (const float*)d_in[4];   // (1,)
    const float* cs = (const float*)d_in[5];   // scalar

    float* ws_phases = (float*)d_ws;                     // B*S floats
    float* ws_freqs  = ws_phases + BATCH * SEQ;          // B*S floats

    float* out_phases = (float*)d_out;                   // (B, S)
    float* out_r      = out_phases + BATCH * SEQ;        // (B,)

    // 512 tiles of 32 tokens, 8 waves (256 thr) per block -> 64 blocks
    const int tiles  = (BATCH * SEQ) / 32;
    const int blocks = tiles / 8;
    kuramoto_proj_wmma<<<blocks, 256, 0, stream>>>(h, Wp, bp, Wf, bf,
                                                   ws_phases, ws_freqs);

    kuramoto_dynamics<<<BATCH, 512, 0, stream>>>(ws_phases, ws_freqs, cs,
                                                 out_phases, out_r);
}